// GCN_12592844112334
// MI455X (gfx1250) — compile-verified
//
#include <hip/hip_runtime.h>
#include <hip/hip_bf16.h>

typedef __attribute__((ext_vector_type(2))) float v2f;
typedef __attribute__((ext_vector_type(8))) float v8f;

#define D_IN  512
#define D_HID 16
#define D_OUT 7
#define LDSB_STRIDE 516   // 516 % 64 == 4 -> conflict-free b64 B reads across 32 lanes

// ---------------- init: deg=1 (self loop), agg1=0, out=0 ----------------
__global__ void k_init(float* __restrict__ deg, float* __restrict__ agg1,
                       float* __restrict__ out, int N) {
    int i = blockIdx.x * blockDim.x + threadIdx.x;
    if (i < N) deg[i] = 1.0f;                 // self-loop contributes 1 to every degree
    if (i < N * D_HID) agg1[i] = 0.0f;
    if (i < N * D_OUT) out[i] = 0.0f;
}

// ---------------- degree count over real edges ----------------
__global__ void k_deg(const long long* __restrict__ dst, float* __restrict__ deg, int E) {
    int e = blockIdx.x * blockDim.x + threadIdx.x;
    if (e >= E) return;
    unsafeAtomicAdd(&deg[(int)dst[e]], 1.0f);
}

// ---------------- deg -> dinv = rsqrt(deg) in place (deg >= 1 always) ----------------
__global__ void k_rsqrt(float* __restrict__ deg, int N) {
    int i = blockIdx.x * blockDim.x + threadIdx.x;
    if (i < N) deg[i] = rsqrtf(deg[i]);
}

// ---------------- h1 = x @ W1 via V_WMMA_F32_16X16X4_F32 ----------------
// One wave computes a 16x16 output tile; K loop 512 in steps of 4 -> 128 wmma/tile.
// A (16x4 f32): lane m=l&15 holds row m; lanes 0-15 -> K=k+0,k+1; lanes 16-31 -> K=k+2,k+3.
// B (4x16 f32): VGPR0 = rows K=k+0 (lanes 0-15) / K=k+2 (lanes 16-31); VGPR1 = K=k+1 / k+3.
// B is read from LDS (W1 staged transposed) so the VMEM pipe is dedicated to streaming x.
__global__ void __launch_bounds__(256)
k_gemm1_wmma(const float* __restrict__ x, const float* __restrict__ W1,
             float* __restrict__ h1, int N) {
    __shared__ float ldsBT[D_HID * LDSB_STRIDE];   // W1 transposed: ldsBT[m][k], ~33KB

    // Cooperative load + transpose of W1 (coalesced global reads, conflict-free LDS writes)
    for (int idx = threadIdx.x; idx < D_IN * D_HID; idx += blockDim.x) {
        int k = idx >> 4;          // 0..511
        int m = idx & 15;          // 0..15
        ldsBT[m * LDSB_STRIDE + k] = W1[idx];
    }
    __syncthreads();

    const int wave  = threadIdx.x >> 5;
    const int lane  = threadIdx.x & 31;
    const int tile  = blockIdx.x * (blockDim.x >> 5) + wave;
    const int tiles = (N + 15) / 16;
    if (tile >= tiles) return;                 // wave-uniform: EXEC stays all-ones below

    const int m     = lane & 15;
    const int khalf = (lane >> 4) * 2;         // 0 or 2

    int row = tile * 16 + m;
    if (row >= N) row = N - 1;                 // clamp loads; store handles the partial tile

    const float* __restrict__ xrow = x + (size_t)row * D_IN;
    const float* __restrict__ brow = &ldsBT[m * LDSB_STRIDE + khalf];

    v8f c = {0.f, 0.f, 0.f, 0.f, 0.f, 0.f, 0.f, 0.f};

    #pragma unroll 8
    for (int k = 0; k < D_IN; k += 4) {
        v2f a = *(const v2f*)(xrow + k + khalf);   // global_load_b64 (stream x)
        v2f b = *(const v2f*)(brow + k);           // ds_load_b64 (LDS pipe)
        c = __builtin_amdgcn_wmma_f32_16x16x4_f32(
                /*neg_a=*/false, a, /*neg_b=*/false, b,
                /*c_mod=*/(short)0, c, /*reuse_a=*/false, /*reuse_b=*/false);
    }

    // C/D layout: c[r] = element (M = r + 8*(lane>>4), Nc = lane&15)
    const int mbase = (lane >> 4) * 8;
    float* __restrict__ orow = h1 + (size_t)tile * 16 * D_HID;
    if (tile * 16 + 16 <= N) {                 // full tile: straight-line stores
        #pragma unroll
        for (int r = 0; r < 8; ++r)
            orow[(mbase + r) * D_HID + m] = c[r];
    } else {                                   // partial last tile: guarded stores
        #pragma unroll
        for (int r = 0; r < 8; ++r)
            if (tile * 16 + mbase + r < N)
                orow[(mbase + r) * D_HID + m] = c[r];
    }
}

// ---------------- layer-1 edge aggregation: agg1[d] += norm * h1[s] ----------------
// 16 threads per edge -> coalesced 64B row gathers and 64B atomic rows (L2-resident).
__global__ void k_aggr1(const long long* __restrict__ src, const long long* __restrict__ dst,
                        const float* __restrict__ dinv, const float* __restrict__ h1,
                        float* __restrict__ agg1, long long total) {
    long long gid = (long long)blockIdx.x * blockDim.x + threadIdx.x;
    if (gid >= total) return;
    int e = (int)(gid >> 4);
    int f = (int)(gid & 15);
    int s = (int)src[e];
    int d = (int)dst[e];
    float nrm = dinv[s] * dinv[d];
    unsafeAtomicAdd(&agg1[d * D_HID + f], nrm * h1[s * D_HID + f]);
}

// ---------------- fused: self-loop + b1 + ReLU + (h @ W2) ----------------
__global__ void k_layer2(const float* __restrict__ dinv, const float* __restrict__ h1,
                         const float* __restrict__ agg1, const float* __restrict__ b1,
                         const float* __restrict__ W2, float* __restrict__ h2, int N) {
    int i = blockIdx.x * blockDim.x + threadIdx.x;
    if (i >= N) return;
    float di2 = dinv[i] * dinv[i];             // self-loop norm = dinv[i]^2
    float h[D_HID];
    #pragma unroll
    for (int k = 0; k < D_HID; ++k) {
        float v = agg1[i * D_HID + k] + di2 * h1[i * D_HID + k] + b1[k];
        h[k] = v > 0.0f ? v : 0.0f;            // ReLU
    }
    #pragma unroll
    for (int j = 0; j < D_OUT; ++j) {
        float acc = 0.0f;
        #pragma unroll
        for (int k = 0; k < D_HID; ++k)
            acc = fmaf(h[k], W2[k * D_OUT + j], acc);
        h2[i * D_OUT + j] = acc;
    }
}

// ---------------- layer-2 edge aggregation into out ----------------
// 8 threads per edge, lane 7 idle.
__global__ void k_aggr2(const long long* __restrict__ src, const long long* __restrict__ dst,
                        const float* __restrict__ dinv, const float* __restrict__ h2,
                        float* __restrict__ out, long long total) {
    long long gid = (long long)blockIdx.x * blockDim.x + threadIdx.x;
    if (gid >= total) return;
    int e = (int)(gid >> 3);
    int c = (int)(gid & 7);
    if (c >= D_OUT) return;
    int s = (int)src[e];
    int d = (int)dst[e];
    float nrm = dinv[s] * dinv[d];
    unsafeAtomicAdd(&out[d * D_OUT + c], nrm * h2[s * D_OUT + c]);
}

// ---------------- fused: self-loop + b2 + log_softmax (in place on out) ----------------
__global__ void k_final(const float* __restrict__ dinv, const float* __restrict__ h2,
                        const float* __restrict__ b2, float* __restrict__ out, int N) {
    int i = blockIdx.x * blockDim.x + threadIdx.x;
    if (i >= N) return;
    float di2 = dinv[i] * dinv[i];
    float v[D_OUT];
    float mx = -3.402823466e+38f;
    #pragma unroll
    for (int c = 0; c < D_OUT; ++c) {
        v[c] = out[i * D_OUT + c] + di2 * h2[i * D_OUT + c] + b2[c];
        mx = fmaxf(mx, v[c]);
    }
    float s = 0.0f;
    #pragma unroll
    for (int c = 0; c < D_OUT; ++c) s += __expf(v[c] - mx);
    float ls = __logf(s);
    #pragma unroll
    for (int c = 0; c < D_OUT; ++c) out[i * D_OUT + c] = v[c] - mx - ls;
}

extern "C" void kernel_launch(void* const* d_in, const int* in_sizes, int n_in,
                              void* d_out, int out_size, void* d_ws, size_t ws_size,
                              hipStream_t stream) {
    const float*     x  = (const float*)d_in[0];
    const long long* ei = (const long long*)d_in[1];
    const float*     W1 = (const float*)d_in[2];
    const float*     b1 = (const float*)d_in[3];
    const float*     W2 = (const float*)d_in[4];
    const float*     b2 = (const float*)d_in[5];
    float*           out = (float*)d_out;

    const int N = in_sizes[0] / D_IN;          // 100000
    const int E = in_sizes[1] / 2;             // 3200000
    const long long* src = ei;
    const long long* dst = ei + E;

    // Workspace carve-up (256B aligned)
    char* ws = (char*)d_ws;
    size_t off = 0;
    auto carve = [&](size_t bytes) -> float* {
        float* p = (float*)(ws + off);
        off += (bytes + 255) & ~(size_t)255;
        return p;
    };
    float* deg  = carve((size_t)N * sizeof(float));            // -> becomes dinv in place
    float* h1   = carve((size_t)N * D_HID * sizeof(float));
    float* agg1 = carve((size_t)N * D_HID * sizeof(float));
    float* h2   = carve((size_t)N * D_OUT * sizeof(float));
    (void)ws_size;

    const int B = 256;

    // 1. init
    {
        int total = N * D_HID;
        k_init<<<(total + B - 1) / B, B, 0, stream>>>(deg, agg1, out, N);
    }
    // 2. degree
    k_deg<<<(E + B - 1) / B, B, 0, stream>>>(dst, deg, E);
    // 3. dinv
    k_rsqrt<<<(N + B - 1) / B, B, 0, stream>>>(deg, N);
    // 4. h1 = x @ W1   (WMMA f32, one wave per 16-row tile, W1 staged in LDS)
    {
        int tiles = (N + 15) / 16;
        int wavesPerBlock = B / 32;
        k_gemm1_wmma<<<(tiles + wavesPerBlock - 1) / wavesPerBlock, B, 0, stream>>>(x, W1, h1, N);
    }
    // 5. layer-1 edge scatter-add
    {
        long long total = (long long)E * D_HID;
        k_aggr1<<<(unsigned)((total + B - 1) / B), B, 0, stream>>>(src, dst, deg, h1, agg1, total);
    }
    // 6. self-loop + bias + relu + small GEMM
    k_layer2<<<(N + B - 1) / B, B, 0, stream>>>(deg, h1, agg1, b1, W2, h2, N);
    // 7. layer-2 edge scatter-add
    {
        long long total = (long long)E * 8;
        k_aggr2<<<(unsigned)((total + B - 1) / B), B, 0, stream>>>(src, dst, deg, h2, out, total);
    }
    // 8. self-loop + b2 + log_softmax
    k_final<<<(N + B - 1) / B, B, 0, stream>>>(deg, h2, b2, out, N);
}